// S4D_30794915512668
// MI455X (gfx1250) — compile-verified
//
#include <hip/hip_runtime.h>
#include <math.h>

// Problem constants (match reference)
#define H_     512
#define NMODE  32      // N/2 complex modes
#define B_     8
#define L_     2048
#define Q_     64      // chunk length
#define NCHUNK 32      // L_/Q_
#define CPITCH 34      // LDS pitch for 32-wide Cout rows (bank-conflict-free)
#define WPITCH 516     // LDS pitch for 512-wide W rows (bank-conflict-free)

typedef __attribute__((ext_vector_type(2))) float v2f;
typedef __attribute__((ext_vector_type(8))) float v8f;

// V_WMMA_F32_16X16X4_F32 : D(16x16 f32) = A(16x4 f32) * B(4x16 f32) + C
__device__ __forceinline__ v8f wmma4(v2f a, v2f b, v8f c) {
  return __builtin_amdgcn_wmma_f32_16x16x4_f32(false, a, false, b, (short)0, c,
                                               false, false);
}

// ---------------------------------------------------------------------------
// Stage 0: per-head mode parameters.
//   w   = exp(dt*A), Ck = C*(w-1)/A
//   Ktap2[h][0..63]   = 0                      (zero pad -> branch-free Toeplitz)
//   Ktap2[h][64+d]    = 2*Re(sum_k Ck_k w_k^d)  d in [0,64)
//   CoutR[h][j][k] =  2*Re(Ck_k w_k^{j+1}),  CoutI[h][j][k] = -2*Im(Ck_k w_k^{j+1})
// one wave (32 lanes = 32 modes) per head
// ---------------------------------------------------------------------------
__global__ void s4d_setup(const float* __restrict__ log_dt,
                          const float* __restrict__ C,
                          const float* __restrict__ log_A_real,
                          const float* __restrict__ A_imag,
                          float* __restrict__ Ktap2,
                          float* __restrict__ CoutR,
                          float* __restrict__ CoutI,
                          float* __restrict__ Wr,
                          float* __restrict__ Wi) {
  const int h = blockIdx.x;
  const int k = threadIdx.x;   // mode, 0..31 (wave32)
  const float dt = expf(log_dt[h]);
  const float Ar = -expf(log_A_real[h * NMODE + k]);
  const float Ai = A_imag[h * NMODE + k];
  const float Ccr = C[(h * NMODE + k) * 2 + 0];
  const float Cci = C[(h * NMODE + k) * 2 + 1];
  const float er = expf(Ar * dt);
  const float wr = er * cosf(Ai * dt);
  const float wi = er * sinf(Ai * dt);
  // e = w - 1 ;  Ck = Cc * e / A  (complex)
  const float exr = wr - 1.0f, exi = wi;
  const float den = Ar * Ar + Ai * Ai;
  const float qr = (exr * Ar + exi * Ai) / den;
  const float qi = (exi * Ar - exr * Ai) / den;
  const float Ckr = Ccr * qr - Cci * qi;
  const float Cki = Ccr * qi + Cci * qr;
  Wr[h * NMODE + k] = wr;
  Wi[h * NMODE + k] = wi;

  // zero pad region
  Ktap2[h * 128 + k] = 0.0f;
  Ktap2[h * 128 + 32 + k] = 0.0f;

  // Toeplitz taps: p = Ck * w^d, reduce 2*Re(p) over lanes
  float pr = Ckr, pi = Cki;
  for (int d = 0; d < Q_; ++d) {
    float v = 2.0f * pr;
    for (int m = 16; m >= 1; m >>= 1) v += __shfl_xor(v, m, 32);
    if (k == 0) Ktap2[h * 128 + 64 + d] = v;
    const float nr = pr * wr - pi * wi;
    const float ni = pr * wi + pi * wr;
    pr = nr; pi = ni;
  }
  // Output projection of carried state: s = Ck * w^{j+1}
  float sr = Ckr * wr - Cki * wi;
  float si = Ckr * wi + Cki * wr;
  for (int j = 0; j < Q_; ++j) {
    CoutR[((size_t)h * Q_ + j) * NMODE + k] = 2.0f * sr;
    CoutI[((size_t)h * Q_ + j) * NMODE + k] = -2.0f * si;
    const float nr = sr * wr - si * wi;
    const float ni = sr * wi + si * wr;
    sr = nr; si = ni;
  }
}

// ---------------------------------------------------------------------------
// Stage 1: mode-state scan. One wave per (b,h); lane = mode. Horner:
//   s <- s*w + u[t]   (== s_c = w^Q s_{c-1} + sum_t w^{Q-1-t} u[cQ+t])
// Stores the state *entering* each chunk to XR/XI (s_{-1} = 0).
// ---------------------------------------------------------------------------
__global__ void s4d_scan(const float* __restrict__ u,
                         const float* __restrict__ Wr,
                         const float* __restrict__ Wi,
                         float* __restrict__ XR,
                         float* __restrict__ XI) {
  const int wid  = (int)((blockIdx.x * blockDim.x + threadIdx.x) >> 5);
  const int lane = threadIdx.x & 31;
  const int b = wid >> 9;        // / H_
  const int h = wid & (H_ - 1);
  const float wr = Wr[h * NMODE + lane];
  const float wi = Wi[h * NMODE + lane];
  float sr = 0.0f, si = 0.0f;
  const float* up = u + (size_t)(b * H_ + h) * L_;
  const size_t xb = ((size_t)(b * H_ + h)) * NCHUNK * NMODE;
  for (int c = 0; c < NCHUNK; ++c) {
    XR[xb + (size_t)c * NMODE + lane] = sr;   // state entering chunk c
    XI[xb + (size_t)c * NMODE + lane] = si;
    for (int t = 0; t < Q_; ++t) {
      const float uu = up[c * Q_ + t];        // wave-uniform address -> 1 request
      const float nr = sr * wr - si * wi + uu;
      const float ni = sr * wi + si * wr;
      sr = nr; si = ni;
    }
  }
}

// ---------------------------------------------------------------------------
// Stage 2: chunked convolution via WMMA + skip + erf-GELU.
// Per head h: Y(64 x 256) = Toeplitz(64x64) @ Uchunks(64x256)
//                         + CoutR(64x32) @ XR(32x256) + CoutI(64x32) @ XI(32x256)
// A-side operands staged in LDS (shared by all 8 waves), branch-free via
// zero-padded taps. Block = 8 waves = 4 row-tiles x 2 col-tiles (64x32 out).
// Grid = H * 8 col-groups.
// ---------------------------------------------------------------------------
__global__ void s4d_conv(const float* __restrict__ u,
                         const float* __restrict__ Ktap2,
                         const float* __restrict__ CoutR,
                         const float* __restrict__ CoutI,
                         const float* __restrict__ XR,
                         const float* __restrict__ XI,
                         const float* __restrict__ Dvec,
                         float* __restrict__ ygelu) {
  __shared__ float sKt[128];
  __shared__ float sCR[Q_ * CPITCH];
  __shared__ float sCI[Q_ * CPITCH];

  const int h      = blockIdx.x >> 3;
  const int g      = blockIdx.x & 7;
  const int waveId = threadIdx.x >> 5;
  const int lane   = threadIdx.x & 31;
  const int half   = lane >> 4;
  const int lm     = lane & 15;
  const int jt     = waveId & 3;                 // row-tile (j = jt*16 ..)
  const int ct     = waveId >> 2;                // col sub-tile
  const int q      = g * 32 + ct * 16 + lm;      // column 0..255
  const int b      = q >> 5;
  const int c      = q & 31;

  // cooperative LDS staging (coalesced global reads)
  if (threadIdx.x < 128) sKt[threadIdx.x] = Ktap2[(size_t)h * 128 + threadIdx.x];
  for (int i = threadIdx.x; i < Q_ * NMODE; i += 256) {
    const int j = i >> 5, k = i & 31;
    sCR[j * CPITCH + k] = CoutR[((size_t)h * Q_ + j) * NMODE + k];
    sCI[j * CPITCH + k] = CoutI[((size_t)h * Q_ + j) * NMODE + k];
  }
  __syncthreads();

  const size_t ub = (size_t)(b * H_ + h) * L_ + (size_t)c * Q_;
  const int jrow = jt * 16 + lm;                 // A-fragment row

  v8f acc = {0.f, 0.f, 0.f, 0.f, 0.f, 0.f, 0.f, 0.f};

  // intra-chunk: lower-triangular Toeplitz, A[j][t] = Ktap2[64 + j - t] (branch-free)
#pragma unroll
  for (int k0 = 0; k0 < Q_; k0 += 4) {
    const int ta = k0 + 2 * half;
    v2f a, bb;
    a.x = sKt[64 + jrow - ta];          // A[jrow][ta]
    a.y = sKt[63 + jrow - ta];          // A[jrow][ta+1]
    bb.x = u[ub + ta];
    bb.y = u[ub + ta + 1];
    acc = wmma4(a, bb, acc);
  }
  // inter-chunk: CoutR @ XR  (K=32)
  const size_t xrb = ((size_t)(b * H_ + h) * NCHUNK + c) * NMODE;
#pragma unroll
  for (int k0 = 0; k0 < NMODE; k0 += 4) {
    const int ka = k0 + 2 * half;
    v2f a, bb;
    a.x = sCR[jrow * CPITCH + ka];
    a.y = sCR[jrow * CPITCH + ka + 1];
    bb.x = XR[xrb + ka];
    bb.y = XR[xrb + ka + 1];
    acc = wmma4(a, bb, acc);
  }
  // inter-chunk: CoutI @ XI  (K=32)
#pragma unroll
  for (int k0 = 0; k0 < NMODE; k0 += 4) {
    const int ka = k0 + 2 * half;
    v2f a, bb;
    a.x = sCI[jrow * CPITCH + ka];
    a.y = sCI[jrow * CPITCH + ka + 1];
    bb.x = XI[xrb + ka];
    bb.y = XI[xrb + ka + 1];
    acc = wmma4(a, bb, acc);
  }
  // epilogue: + D*u, erf-GELU
  const float Dh = Dvec[h];
#pragma unroll
  for (int v = 0; v < 8; ++v) {
    const int j = jt * 16 + v + 8 * half;                 // D row = v + 8*half
    const size_t idx = (size_t)(b * H_ + h) * L_ + (size_t)c * Q_ + j;
    const float uu = u[idx];
    const float y = acc[v] + Dh * uu;
    ygelu[idx] = 0.5f * y * (1.0f + erff(y * 0.70710678118654752f));
  }
}

// ---------------------------------------------------------------------------
// Stage 3: 1x1 conv GEMM (M=2H, K=H, N=B*L) + bias + GLU, f32 WMMA.
// W tile (16 'a' rows + 16 'g' rows, K=512) staged in dynamic LDS, shared by
// all 8 waves of the block; B-fragments shared between the a- and g-WMMAs.
// Grid: 32 m-tiles x 1024 n-tiles = 4096 blocks x 8 waves.
// ---------------------------------------------------------------------------
__global__ void s4d_glu(const float* __restrict__ ygelu,
                        const float* __restrict__ W,
                        const float* __restrict__ bias,
                        float* __restrict__ out) {
  extern __shared__ float sW[];                  // 32 rows x WPITCH
  const int bid    = blockIdx.x;
  const int waveId = threadIdx.x >> 5;
  const int lane   = threadIdx.x & 31;
  const int half   = lane >> 4;
  const int lm     = lane & 15;
  const int mtile  = bid >> 7;                    // 0..31
  const int ntile  = (bid & 127) * 8 + waveId;    // 0..1023
  const int col    = ntile * 16 + lm;             // 0..16383
  const int bb     = col >> 11;                   // batch
  const int l      = col & (L_ - 1);

  // stage W tile: rows 0..15 = a-rows (h), rows 16..31 = g-rows (H+h)
  for (int i = threadIdx.x; i < (32 * H_) / 4; i += 256) {
    const int r = (i * 4) >> 9;                   // row 0..31
    const int k = (i * 4) & (H_ - 1);
    const int src = (r < 16) ? (mtile * 16 + r) : (H_ + mtile * 16 + (r - 16));
    const float4 v4 = *(const float4*)(W + (size_t)src * H_ + k);
    float* dst = sW + r * WPITCH + k;
    dst[0] = v4.x; dst[1] = v4.y; dst[2] = v4.z; dst[3] = v4.w;
  }
  __syncthreads();

  const float* yb = ygelu + (size_t)bb * H_ * L_ + l;
  const float* la = sW + lm * WPITCH;             // a-row for this lane
  const float* lg = sW + (16 + lm) * WPITCH;      // g-row for this lane

  v8f accA = {0.f, 0.f, 0.f, 0.f, 0.f, 0.f, 0.f, 0.f};
  v8f accG = {0.f, 0.f, 0.f, 0.f, 0.f, 0.f, 0.f, 0.f};

#pragma unroll 4
  for (int k0 = 0; k0 < H_; k0 += 4) {
    const int ka = k0 + 2 * half;
    v2f bf, aA, aG;
    bf.x = yb[(size_t)ka * L_];
    bf.y = yb[(size_t)(ka + 1) * L_];
    aA.x = la[ka];  aA.y = la[ka + 1];
    aG.x = lg[ka];  aG.y = lg[ka + 1];
    if (k0 + 32 < H_) {
      __builtin_prefetch(yb + (size_t)(k0 + 32) * L_, 0, 0);  // global_prefetch_b8
    }
    accA = wmma4(aA, bf, accA);
    accG = wmma4(aG, bf, accG);
  }
#pragma unroll
  for (int v = 0; v < 8; ++v) {
    const int hr = mtile * 16 + v + 8 * half;
    const float a = accA[v] + bias[hr];
    const float gv = accG[v] + bias[H_ + hr];
    out[((size_t)(bb * H_ + hr)) * L_ + l] = a * (1.0f / (1.0f + expf(-gv)));
  }
}

// ---------------------------------------------------------------------------
// Workspace layout (floats):
//   Ktap2 : H*128                 =    65536
//   CoutR : H*64*32               =  1048576
//   CoutI : H*64*32               =  1048576
//   Wr    : H*32                  =    16384
//   Wi    : H*32                  =    16384
//   XR    : B*H*32*32             =  4194304
//   XI    : B*H*32*32             =  4194304
//   ygelu : B*H*L                 =  8388608
// total ~19.0M floats (~76 MB)
// ---------------------------------------------------------------------------
extern "C" void kernel_launch(void* const* d_in, const int* in_sizes, int n_in,
                              void* d_out, int out_size, void* d_ws, size_t ws_size,
                              hipStream_t stream) {
  (void)in_sizes; (void)n_in; (void)out_size; (void)ws_size;
  const float* u          = (const float*)d_in[0];
  const float* log_dt     = (const float*)d_in[1];
  const float* C          = (const float*)d_in[2];
  const float* log_A_real = (const float*)d_in[3];
  const float* A_imag     = (const float*)d_in[4];
  const float* Dvec       = (const float*)d_in[5];
  const float* W          = (const float*)d_in[6];
  const float* bias       = (const float*)d_in[7];
  float* out = (float*)d_out;

  float* ws = (float*)d_ws;
  float* Ktap2 = ws;                          // H*128
  float* CoutR = Ktap2 + (size_t)H_ * 128;
  float* CoutI = CoutR + (size_t)H_ * Q_ * NMODE;
  float* Wr    = CoutI + (size_t)H_ * Q_ * NMODE;
  float* Wi    = Wr    + (size_t)H_ * NMODE;
  float* XR    = Wi    + (size_t)H_ * NMODE;
  float* XI    = XR    + (size_t)B_ * H_ * NCHUNK * NMODE;
  float* ygelu = XI    + (size_t)B_ * H_ * NCHUNK * NMODE;

  s4d_setup<<<H_, 32, 0, stream>>>(log_dt, C, log_A_real, A_imag,
                                   Ktap2, CoutR, CoutI, Wr, Wi);
  s4d_scan<<<(B_ * H_) / 8, 256, 0, stream>>>(u, Wr, Wi, XR, XI);
  s4d_conv<<<H_ * 8, 256, 0, stream>>>(u, Ktap2, CoutR, CoutI, XR, XI, Dvec, ygelu);
  s4d_glu<<<4096, 256, (size_t)(32 * WPITCH) * sizeof(float), stream>>>(ygelu, W,
                                                                        bias, out);
}